// GAT_12257836663204
// MI455X (gfx1250) — compile-verified
//
#include <hip/hip_runtime.h>

typedef __attribute__((ext_vector_type(2))) float v2f;
typedef __attribute__((ext_vector_type(8))) float v8f;

static __device__ __forceinline__ unsigned mapOrderedF32(float f) {
  unsigned u = __float_as_uint(f);
  return (u & 0x80000000u) ? ~u : (u | 0x80000000u);
}
static __device__ __forceinline__ float unmapOrderedF32(unsigned u) {
  return (u & 0x80000000u) ? __uint_as_float(u & 0x7FFFFFFFu) : __uint_as_float(~u);
}

// ---------------- elementwise helpers ----------------
__global__ void k_zero(float* p, long n) {
  long i = (long)blockIdx.x * blockDim.x + threadIdx.x;
  if (i < n) p[i] = 0.f;
}

__global__ void k_edge_deg(const int* __restrict__ ei, const float* __restrict__ eattr,
                           float* __restrict__ deg, float* __restrict__ lsum, int E) {
  int e = blockIdx.x * blockDim.x + threadIdx.x;
  if (e >= E) return;
  int dst = ei[E + e];
  atomicAdd(&deg[dst], 1.f);
  const float* ea = eattr + (long)e * 8;
  float* ls = lsum + (long)dst * 8;
  #pragma unroll
  for (int d = 0; d < 8; ++d) atomicAdd(&ls[d], ea[d]);
}

__global__ void k_loop_div(float* lsum, const float* __restrict__ deg, int N) {
  int i = blockIdx.x * blockDim.x + threadIdx.x;
  if (i >= N * 8) return;
  lsum[i] /= fmaxf(deg[i >> 3], 1.f);
}

__global__ void k_concat(const float* __restrict__ x, const float* __restrict__ h,
                         float* __restrict__ gi, int N) {
  long i = (long)blockIdx.x * blockDim.x + threadIdx.x;
  if (i >= (long)N * 80) return;
  int n = (int)(i / 80), c = (int)(i % 80);
  gi[i] = (c < 16) ? x[(long)n * 16 + c] : h[(long)n * 64 + (c - 16)];
}

// ---------------- WMMA f32 GEMM: C[M,NC] = act(A[M,K] @ W[K,NC] + bias) ----------------
// Compile-time K, NC. Block = 256 threads = 8 waves; each wave owns one 16-row strip
// and all NC columns (NT = NC/16 accumulator tiles). W staged in LDS in pair-interleaved
// layout so each WMMA B-fragment is one aligned ds_load_b64.
template <int K, int NC>
__global__ __launch_bounds__(256) void k_wmma_gemm(const float* __restrict__ A,
                                                   const float* __restrict__ W,
                                                   const float* __restrict__ bias,
                                                   float* __restrict__ C, int M, int relu) {
  constexpr int NT = NC / 16;
  constexpr int PSTRIDE = 2 * NC + 2;  // floats per k-pair region (+2 pad de-phases banks)
  __shared__ float ldsB[(K / 2) * PSTRIDE];

  // cooperative stage of W[K][NC] into pair-interleaved LDS layout
  for (int idx = threadIdx.x; idx < K * NC; idx += 256) {
    int k = idx / NC, n = idx % NC;
    ldsB[(k >> 1) * PSTRIDE + n * 2 + (k & 1)] = W[idx];
  }
  __syncthreads();

  int strip = blockIdx.x * 8 + (threadIdx.x >> 5);
  if (strip >= (M >> 4)) return;            // wave-uniform -> EXEC stays all-ones
  int lane = threadIdx.x & 31;
  int half = lane >> 4, l16 = lane & 15;

  // A fragment base: lane covers row l16, k-locals 2*half .. 2*half+1  (8B aligned)
  const float* pA = A + (long)(strip * 16 + l16) * K + 2 * half;

  v8f acc[NT];
  const v8f vzero = {0.f, 0.f, 0.f, 0.f, 0.f, 0.f, 0.f, 0.f};
  #pragma unroll
  for (int t = 0; t < NT; ++t) acc[t] = vzero;

  #pragma unroll
  for (int k = 0; k < K; k += 4) {
    v2f a = *(const v2f*)(pA + k);                       // global_load_b64, imm offset
    const float* bp = &ldsB[(k / 2 + half) * PSTRIDE + l16 * 2];
    #pragma unroll
    for (int t = 0; t < NT; ++t) {
      v2f b = *(const v2f*)(bp + t * 32);                // ds_load_b64, imm offset
      acc[t] = __builtin_amdgcn_wmma_f32_16x16x4_f32(false, a, false, b, (short)0,
                                                     acc[t], false, false);
    }
  }

  #pragma unroll
  for (int t = 0; t < NT; ++t) {
    int col = t * 16 + l16;
    float bv = bias ? bias[col] : 0.f;
    #pragma unroll
    for (int r = 0; r < 8; ++r) {
      int row = strip * 16 + r + half * 8;
      float v = acc[t][r] + bv;
      if (relu) v = fmaxf(v, 0.f);
      C[(long)row * NC + col] = v;
    }
  }
}

// ---------------- GAT attention pipeline ----------------
__global__ void k_attcoef(const float* __restrict__ xh, const float* __restrict__ asrc,
                          const float* __restrict__ adst, float* __restrict__ a_s,
                          float* __restrict__ a_d, int N) {
  int i = blockIdx.x * blockDim.x + threadIdx.x;
  if (i >= N * 4) return;
  int n = i >> 2, hh = i & 3;
  const float* xr = xh + (long)n * 64 + hh * 16;
  float s = 0.f, d = 0.f;
  #pragma unroll
  for (int c = 0; c < 16; ++c) {
    float v = xr[c];
    s += v * asrc[hh * 16 + c];
    d += v * adst[hh * 16 + c];
  }
  a_s[i] = s;
  a_d[i] = d;
}

// fold edge_w[8,64] with att_edge[4,16] -> eaw[8,4]
__global__ void k_eaw(const float* __restrict__ ew, const float* __restrict__ ae,
                      float* __restrict__ eaw) {
  int i = threadIdx.x;
  if (i >= 32) return;
  int d = i >> 2, hh = i & 3;
  float s = 0.f;
  #pragma unroll
  for (int c = 0; c < 16; ++c) s += ew[d * 64 + hh * 16 + c] * ae[hh * 16 + c];
  eaw[i] = s;
}

__global__ void k_alpha_max(const int* __restrict__ ei, const float* __restrict__ eattr,
                            const float* __restrict__ lattr, const float* __restrict__ a_s,
                            const float* __restrict__ a_d, const float* __restrict__ eaw,
                            float* __restrict__ alpha, unsigned* __restrict__ mMap,
                            int E, int N) {
  long i = (long)blockIdx.x * blockDim.x + threadIdx.x;
  if (i >= (long)(E + N) * 4) return;
  long e = i >> 2;
  int hh = (int)(i & 3);
  int s, dst;
  const float* ea;
  if (e < E) { s = ei[e]; dst = ei[E + e]; ea = eattr + e * 8; }
  else       { s = dst = (int)(e - E);     ea = lattr + (e - E) * 8; }
  float a = a_s[(long)s * 4 + hh] + a_d[(long)dst * 4 + hh];
  #pragma unroll
  for (int d = 0; d < 8; ++d) a += ea[d] * eaw[d * 4 + hh];
  a = (a > 0.f) ? a : 0.2f * a;  // leaky_relu 0.2
  alpha[i] = a;
  atomicMax(&mMap[(long)dst * 4 + hh], mapOrderedF32(a));
}

__global__ void k_expdenom(const int* __restrict__ ei, const unsigned* __restrict__ mMap,
                           float* __restrict__ alpha, float* __restrict__ denom,
                           int E, int N) {
  long i = (long)blockIdx.x * blockDim.x + threadIdx.x;
  if (i >= (long)(E + N) * 4) return;
  long e = i >> 2;
  int hh = (int)(i & 3);
  int dst = (e < E) ? ei[E + e] : (int)(e - E);
  float mm = unmapOrderedF32(mMap[(long)dst * 4 + hh]);
  float ex = expf(alpha[i] - mm);
  alpha[i] = ex;
  atomicAdd(&denom[(long)dst * 4 + hh], ex);
}

__global__ void k_scatter(const int* __restrict__ ei, const float* __restrict__ alpha,
                          const float* __restrict__ denom, const float* __restrict__ xh,
                          float* __restrict__ acc, int E, int N) {
  long i = (long)blockIdx.x * blockDim.x + threadIdx.x;
  if (i >= (long)(E + N) * 4) return;
  long e = i >> 2;
  int hh = (int)(i & 3);
  int s, dst;
  if (e < E) { s = ei[e]; dst = ei[E + e]; }
  else       { s = dst = (int)(e - E); }
  float w = alpha[i] / (denom[(long)dst * 4 + hh] + 1e-16f);
  const float* xr = xh + (long)s * 64 + hh * 16;
  float* ar = acc + (long)dst * 64 + hh * 16;
  #pragma unroll
  for (int c = 0; c < 16; ++c) atomicAdd(&ar[c], xr[c] * w);
}

// out+bias -> layernorm -> elu -> h
__global__ void k_layerfin(const float* __restrict__ acc, const float* __restrict__ bias,
                           const float* __restrict__ g, const float* __restrict__ b,
                           float* __restrict__ h, int N) {
  int n = blockIdx.x * blockDim.x + threadIdx.x;
  if (n >= N) return;
  const float* r = acc + (long)n * 64;
  float mu = 0.f;
  for (int c = 0; c < 64; ++c) mu += r[c] + bias[c];
  mu *= (1.f / 64.f);
  float var = 0.f;
  for (int c = 0; c < 64; ++c) { float d = r[c] + bias[c] - mu; var += d * d; }
  var *= (1.f / 64.f);
  float inv = rsqrtf(var + 1e-5f);
  float* hr = h + (long)n * 64;
  for (int c = 0; c < 64; ++c) {
    float y = (r[c] + bias[c] - mu) * inv * g[c] + b[c];
    hr[c] = (y > 0.f) ? y : expm1f(y);  // elu
  }
}

__global__ void k_layernorm(float* __restrict__ hbuf, const float* __restrict__ g,
                            const float* __restrict__ b, int N) {
  int n = blockIdx.x * blockDim.x + threadIdx.x;
  if (n >= N) return;
  float* r = hbuf + (long)n * 64;
  float mu = 0.f;
  for (int c = 0; c < 64; ++c) mu += r[c];
  mu *= (1.f / 64.f);
  float var = 0.f;
  for (int c = 0; c < 64; ++c) { float d = r[c] - mu; var += d * d; }
  var *= (1.f / 64.f);
  float inv = rsqrtf(var + 1e-5f);
  for (int c = 0; c < 64; ++c) r[c] = (r[c] - mu) * inv * g[c] + b[c];
}

__global__ void k_pool(const float* __restrict__ g, const int* __restrict__ batch,
                       float* __restrict__ gemb, float* __restrict__ cnt, int N) {
  long i = (long)blockIdx.x * blockDim.x + threadIdx.x;
  if (i >= (long)64 * N) return;
  int n = (int)(i >> 6), c = (int)(i & 63);
  int b = batch[n];
  atomicAdd(&gemb[(long)b * 64 + c], g[i]);
  if (c == 0) atomicAdd(&cnt[b], 1.f);
}

__global__ void k_pooldiv(float* gemb, const float* __restrict__ cnt, int B) {
  int i = blockIdx.x * blockDim.x + threadIdx.x;
  if (i >= B * 64) return;
  gemb[i] /= fmaxf(cnt[i >> 6], 1.f);
}

// node decoder tail: 32 -> 8 -> 1
__global__ void k_mlptail(const float* __restrict__ z1, const float* __restrict__ w2,
                          const float* __restrict__ b2, const float* __restrict__ w3,
                          const float* __restrict__ b3, float* __restrict__ outv, int N) {
  int n = blockIdx.x * blockDim.x + threadIdx.x;
  if (n >= N) return;
  const float* zr = z1 + (long)n * 32;
  float z[32];
  #pragma unroll
  for (int k = 0; k < 32; ++k) z[k] = zr[k];
  float acc = b3[0];
  #pragma unroll
  for (int j = 0; j < 8; ++j) {
    float a = b2[j];
    #pragma unroll
    for (int k = 0; k < 32; ++k) a += z[k] * w2[k * 8 + j];
    acc += fmaxf(a, 0.f) * w3[j];
  }
  outv[n] = acc;
}

// graph decoder: 64 -> 32 -> 8 -> 1, one thread per graph
__global__ void k_gmlp(const float* __restrict__ gemb, const float* __restrict__ w1,
                       const float* __restrict__ b1, const float* __restrict__ w2,
                       const float* __restrict__ b2, const float* __restrict__ w3,
                       const float* __restrict__ b3, float* __restrict__ outv, int B) {
  int bb = blockIdx.x * blockDim.x + threadIdx.x;
  if (bb >= B) return;
  const float* gr = gemb + (long)bb * 64;
  float z1[32];
  #pragma unroll
  for (int j = 0; j < 32; ++j) {
    float a = b1[j];
    for (int k = 0; k < 64; ++k) a += gr[k] * w1[k * 32 + j];
    z1[j] = fmaxf(a, 0.f);
  }
  float acc = b3[0];
  #pragma unroll
  for (int j = 0; j < 8; ++j) {
    float a = b2[j];
    #pragma unroll
    for (int k = 0; k < 32; ++k) a += z1[k] * w2[k * 8 + j];
    acc += fmaxf(a, 0.f) * w3[j];
  }
  outv[bb] = acc;
}

extern "C" void kernel_launch(void* const* d_in, const int* in_sizes, int n_in,
                              void* d_out, int out_size, void* d_ws, size_t ws_size,
                              hipStream_t stream) {
  (void)n_in; (void)ws_size;
  const float* x     = (const float*)d_in[0];
  const int*   ei    = (const int*)  d_in[1];
  const float* eattr = (const float*)d_in[2];
  const int*   batch = (const int*)  d_in[3];
  const float* niw   = (const float*)d_in[4];
  const float* nib   = (const float*)d_in[5];
  const float* glinw = (const float*)d_in[6];
  const float* gasrc = (const float*)d_in[7];
  const float* gadst = (const float*)d_in[8];
  const float* gew   = (const float*)d_in[9];
  const float* gae   = (const float*)d_in[10];
  const float* gbias = (const float*)d_in[11];
  const float* glng  = (const float*)d_in[12];
  const float* glnb  = (const float*)d_in[13];
  const float* now_  = (const float*)d_in[14];
  const float* nob   = (const float*)d_in[15];
  const float* olng  = (const float*)d_in[16];
  const float* olnb  = (const float*)d_in[17];
  const float* giw   = (const float*)d_in[18];
  const float* gib   = (const float*)d_in[19];
  const float* grlng = (const float*)d_in[20];
  const float* grlnb = (const float*)d_in[21];
  const float* dnw1  = (const float*)d_in[22];
  const float* dnb1  = (const float*)d_in[23];
  const float* dnw2  = (const float*)d_in[24];
  const float* dnb2  = (const float*)d_in[25];
  const float* dnw3  = (const float*)d_in[26];
  const float* dnb3  = (const float*)d_in[27];
  const float* dgw1  = (const float*)d_in[28];
  const float* dgb1  = (const float*)d_in[29];
  const float* dgw2  = (const float*)d_in[30];
  const float* dgb2  = (const float*)d_in[31];
  const float* dgw3  = (const float*)d_in[32];
  const float* dgb3  = (const float*)d_in[33];

  const int N = in_sizes[0] / 16;     // 100000 (multiple of 16)
  const int E = in_sizes[1] / 2;      // 1600000
  const int B = out_size - N;         // 64
  float* outp = (float*)d_out;

  // workspace layout (floats)
  float* ws = (float*)d_ws;
  long o = 0;
  float* deg   = ws + o; o += N;
  float* lattr = ws + o; o += (long)8 * N;
  float* h     = ws + o; o += (long)64 * N;
  float* gi    = ws + o; o += (long)80 * N;   // also reused as g
  float* xh    = ws + o; o += (long)64 * N;   // also reused as node_emb
  float* a_s   = ws + o; o += (long)4 * N;
  float* a_d   = ws + o; o += (long)4 * N;
  float* alpha = ws + o; o += (long)4 * (E + N);
  float* mMapF = ws + o; o += (long)4 * N;    // used as unsigned (contiguous with denom+accum)
  float* denom = ws + o; o += (long)4 * N;
  float* accum = ws + o; o += (long)64 * N;   // also reused as z1
  float* gemb  = ws + o; o += (long)64 * B;
  float* cnt   = ws + o; o += B;
  float* eaw   = ws + o; o += 32;

  const int T = 256;
  auto cdiv = [](long a, long b) { return (unsigned)((a + b - 1) / b); };
  const unsigned gemmBlocks = cdiv(N / 16, 8);  // 8 waves/block, strip per wave

  // self-loop attr: deg + incoming-mean
  k_zero<<<cdiv((long)9 * N, T), T, 0, stream>>>(deg, (long)9 * N);
  k_edge_deg<<<cdiv(E, T), T, 0, stream>>>(ei, eattr, deg, lattr, E);
  k_loop_div<<<cdiv((long)8 * N, T), T, 0, stream>>>(lattr, deg, N);

  // h = relu(x @ node_in_w + b)      [N,16]x[16,64]
  k_wmma_gemm<16, 64><<<gemmBlocks, T, 0, stream>>>(x, niw, nib, h, N, 1);

  const long EN4 = (long)(E + N) * 4;
  for (int l = 0; l < 3; ++l) {
    k_concat<<<cdiv((long)80 * N, T), T, 0, stream>>>(x, h, gi, N);
    k_wmma_gemm<80, 64><<<gemmBlocks, T, 0, stream>>>(gi, glinw + (long)l * 80 * 64,
                                                      nullptr, xh, N, 0);
    k_attcoef<<<cdiv((long)4 * N, T), T, 0, stream>>>(xh, gasrc + l * 64, gadst + l * 64,
                                                      a_s, a_d, N);
    k_eaw<<<1, 32, 0, stream>>>(gew + (long)l * 8 * 64, gae + l * 64, eaw);
    // zero mMap(4N) + denom(4N) + accum(64N) in one contiguous fill
    k_zero<<<cdiv((long)72 * N, T), T, 0, stream>>>(mMapF, (long)72 * N);
    k_alpha_max<<<cdiv(EN4, T), T, 0, stream>>>(ei, eattr, lattr, a_s, a_d, eaw, alpha,
                                                (unsigned*)mMapF, E, N);
    k_expdenom<<<cdiv(EN4, T), T, 0, stream>>>(ei, (const unsigned*)mMapF, alpha, denom, E, N);
    k_scatter<<<cdiv(EN4, T), T, 0, stream>>>(ei, alpha, denom, xh, accum, E, N);
    k_layerfin<<<cdiv(N, T), T, 0, stream>>>(accum, gbias + l * 64, glng + l * 64,
                                             glnb + l * 64, h, N);
  }

  // graph path: g = LN(relu(h @ graph_in_w + b)); pool by batch
  k_wmma_gemm<64, 64><<<gemmBlocks, T, 0, stream>>>(h, giw, gib, gi, N, 1);
  k_layernorm<<<cdiv(N, T), T, 0, stream>>>(gi, grlng, grlnb, N);
  k_zero<<<cdiv((long)64 * B + B, T), T, 0, stream>>>(gemb, (long)64 * B + B);
  k_pool<<<cdiv((long)64 * N, T), T, 0, stream>>>(gi, batch, gemb, cnt, N);
  k_pooldiv<<<cdiv((long)64 * B, T), T, 0, stream>>>(gemb, cnt, B);

  // node_emb = LN(relu(h @ node_out_w + b))  (into xh)
  k_wmma_gemm<64, 64><<<gemmBlocks, T, 0, stream>>>(h, now_, nob, xh, N, 1);
  k_layernorm<<<cdiv(N, T), T, 0, stream>>>(xh, olng, olnb, N);

  // node decoder: z1 = relu([x, node_emb] @ dn_w1 + b1) [N,80]x[80,32], then 32->8->1
  k_concat<<<cdiv((long)80 * N, T), T, 0, stream>>>(x, xh, gi, N);
  k_wmma_gemm<80, 32><<<gemmBlocks, T, 0, stream>>>(gi, dnw1, dnb1, accum, N, 1);
  k_mlptail<<<cdiv(N, T), T, 0, stream>>>(accum, dnw2, dnb2, dnw3, dnb3, outp, N);

  // graph decoder
  k_gmlp<<<1, 64, 0, stream>>>(gemb, dgw1, dgb1, dgw2, dgb2, dgw3, dgb3, outp + N, B);
}